// DepthNet_27238682591967
// MI455X (gfx1250) — compile-verified
//
#include <hip/hip_runtime.h>
#include <hip/hip_bf16.h>

// Problem constants from the reference
#define H_    128
#define W_    160
#define D_    32
#define C_    32
#define CD_   8
#define NPIX  (H_ * W_)
#define NV_   3
#define LAMB_ 1.5f

typedef __attribute__((ext_vector_type(2))) float v2f;
typedef __attribute__((ext_vector_type(8))) float v8f;

__device__ __forceinline__ int iclampi(int x, int lo, int hi) {
    return x < lo ? lo : (x > hi ? hi : x);
}

// Standard adjugate 4x4 inverse (row-major)
__device__ void invert4x4(const float* m, float* o) {
    float inv[16];
    inv[0]  =  m[5]*m[10]*m[15] - m[5]*m[11]*m[14] - m[9]*m[6]*m[15] + m[9]*m[7]*m[14] + m[13]*m[6]*m[11] - m[13]*m[7]*m[10];
    inv[4]  = -m[4]*m[10]*m[15] + m[4]*m[11]*m[14] + m[8]*m[6]*m[15] - m[8]*m[7]*m[14] - m[12]*m[6]*m[11] + m[12]*m[7]*m[10];
    inv[8]  =  m[4]*m[9]*m[15]  - m[4]*m[11]*m[13] - m[8]*m[5]*m[15] + m[8]*m[7]*m[13] + m[12]*m[5]*m[11] - m[12]*m[7]*m[9];
    inv[12] = -m[4]*m[9]*m[14]  + m[4]*m[10]*m[13] + m[8]*m[5]*m[14] - m[8]*m[6]*m[13] - m[12]*m[5]*m[10] + m[12]*m[6]*m[9];
    inv[1]  = -m[1]*m[10]*m[15] + m[1]*m[11]*m[14] + m[9]*m[2]*m[15] - m[9]*m[3]*m[14] - m[13]*m[2]*m[11] + m[13]*m[3]*m[10];
    inv[5]  =  m[0]*m[10]*m[15] - m[0]*m[11]*m[14] - m[8]*m[2]*m[15] + m[8]*m[3]*m[14] + m[12]*m[2]*m[11] - m[12]*m[3]*m[10];
    inv[9]  = -m[0]*m[9]*m[15]  + m[0]*m[11]*m[13] + m[8]*m[1]*m[15] - m[8]*m[3]*m[13] - m[12]*m[1]*m[11] + m[12]*m[3]*m[9];
    inv[13] =  m[0]*m[9]*m[14]  - m[0]*m[10]*m[13] - m[8]*m[1]*m[14] + m[8]*m[2]*m[13] + m[12]*m[1]*m[10] - m[12]*m[2]*m[9];
    inv[2]  =  m[1]*m[6]*m[15]  - m[1]*m[7]*m[14]  - m[5]*m[2]*m[15] + m[5]*m[3]*m[14] + m[13]*m[2]*m[7]  - m[13]*m[3]*m[6];
    inv[6]  = -m[0]*m[6]*m[15]  + m[0]*m[7]*m[14]  + m[4]*m[2]*m[15] - m[4]*m[3]*m[14] - m[12]*m[2]*m[7]  + m[12]*m[3]*m[6];
    inv[10] =  m[0]*m[5]*m[15]  - m[0]*m[7]*m[13]  - m[4]*m[1]*m[15] + m[4]*m[3]*m[13] + m[12]*m[1]*m[7]  - m[12]*m[3]*m[5];
    inv[14] = -m[0]*m[5]*m[14]  + m[0]*m[6]*m[13]  + m[4]*m[1]*m[14] - m[4]*m[2]*m[13] - m[12]*m[1]*m[6]  + m[12]*m[2]*m[5];
    inv[3]  = -m[1]*m[6]*m[11]  + m[1]*m[7]*m[10]  + m[5]*m[2]*m[11] - m[5]*m[3]*m[10] - m[9]*m[2]*m[7]   + m[9]*m[3]*m[6];
    inv[7]  =  m[0]*m[6]*m[11]  - m[0]*m[7]*m[10]  - m[4]*m[2]*m[11] + m[4]*m[3]*m[10] + m[8]*m[2]*m[7]   - m[8]*m[3]*m[6];
    inv[11] = -m[0]*m[5]*m[11]  + m[0]*m[7]*m[9]   + m[4]*m[1]*m[11] - m[4]*m[3]*m[9]  - m[8]*m[1]*m[7]   + m[8]*m[3]*m[5];
    inv[15] =  m[0]*m[5]*m[10]  - m[0]*m[6]*m[9]   - m[4]*m[1]*m[10] + m[4]*m[2]*m[9]  + m[8]*m[1]*m[6]   - m[8]*m[2]*m[5];
    float det = m[0]*inv[0] + m[1]*inv[4] + m[2]*inv[8] + m[3]*inv[12];
    det = 1.0f / det;
    for (int i = 0; i < 16; ++i) o[i] = inv[i] * det;
}

// 1 thread: ws[(v-1)*12 + {0..8}] = rot rows of P_v * inv(P_0); ws[.. +9+{0..2}] = tr
__global__ void proj_setup_kernel(const float* __restrict__ proj, float* __restrict__ ws) {
    if (threadIdx.x != 0 || blockIdx.x != 0) return;
    float invP[16];
    invert4x4(proj, invP);
    for (int v = 1; v < NV_; ++v) {
        const float* Pv = proj + 16 * v;
        float* o = ws + (v - 1) * 12;
        for (int r = 0; r < 3; ++r) {
            float m0 = 0.f, m1 = 0.f, m2 = 0.f, m3 = 0.f;
            for (int k = 0; k < 4; ++k) {
                const float p = Pv[r * 4 + k];
                m0 += p * invP[k * 4 + 0];
                m1 += p * invP[k * 4 + 1];
                m2 += p * invP[k * 4 + 2];
                m3 += p * invP[k * 4 + 3];
            }
            o[r * 3 + 0] = m0; o[r * 3 + 1] = m1; o[r * 3 + 2] = m2;
            o[9 + r] = m3;
        }
    }
}

// Wave layout: 16 pixels per wave32. Lane L and L+16 share pixel p = L&15 and split
// each 4-channel chunk as {4j,4j+1} / {4j+2,4j+3} — exactly the K-split of the
// V_WMMA_F32_16X16X4_F32 A/B VGPR layout, so the WMMA performs the full 40-channel
// reg_w dot (including the cross-lane-half K reduction) with no ds_permute needed.
__global__ __launch_bounds__(256) void depthnet_fused_kernel(
    const float* __restrict__ features,      // (V, C, H, W)
    const float* __restrict__ deps,          // (V, CD, H, W)
    const float* __restrict__ depth_values,  // (D, H, W)
    const float* __restrict__ reg_w,         // (40)
    const float* __restrict__ projw,         // 2 x (rot9 + tr3)
    float* __restrict__ out)                 // depth | exp_var | prob
{
    __shared__ float lds_cost[128][33];      // 16.5KB, padded vs bank conflicts

    const int lane       = threadIdx.x & 31;
    const int half       = lane >> 4;        // 0: chans {4j,4j+1}, 1: {4j+2,4j+3}
    const int p16        = lane & 15;
    const int slot       = (threadIdx.x >> 5) * 16 + p16;
    const int waveGlobal = (blockIdx.x * blockDim.x + threadIdx.x) >> 5;
    const int pix        = waveGlobal * 16 + p16;          // 0 .. NPIX-1 (exact fit)
    const int hh         = pix / W_;
    const int ww         = pix - hh * W_;
    const int ofs        = half * 2;

    // Homography: rx = rot * (x, y, 1), per source view
    const float xf = (float)ww, yf = (float)hh;
    float rx[2][3], tr[2][3];
#pragma unroll
    for (int v = 0; v < 2; ++v) {
        const float* m = projw + v * 12;
#pragma unroll
        for (int i = 0; i < 3; ++i) {
            rx[v][i] = m[i * 3 + 0] * xf + m[i * 3 + 1] * yf + m[i * 3 + 2];
            tr[v][i] = m[9 + i];
        }
    }

    // Preload this lane's 20 reg_w coefficients and 20 ref-view channel values.
    float rwv[20], refv[20];
#pragma unroll
    for (int j = 0; j < 10; ++j) {
#pragma unroll
        for (int t = 0; t < 2; ++t) {
            const int c = 4 * j + ofs + t;
            rwv[2 * j + t]  = reg_w[c];
            refv[2 * j + t] = (c < C_) ? features[c * NPIX + pix]
                                       : deps[(c - C_) * NPIX + pix];
        }
    }

    // Rolled depth loop (I-cache friendly); cost staged in LDS.
    for (int d = 0; d < D_; ++d) {
        const float dval = depth_values[d * NPIX + pix];

        float wgt[2][4];
        int   lin[2][4];
#pragma unroll
        for (int v = 0; v < 2; ++v) {
            const float z   = rx[v][2] * dval + tr[v][2];
            const float iz  = 1.0f / z;
            const float px  = (rx[v][0] * dval + tr[v][0]) * iz;
            const float py  = (rx[v][1] * dval + tr[v][1]) * iz;
            const float x0f = floorf(px), y0f = floorf(py);
            const int   x0  = (int)x0f,   y0  = (int)y0f;
            const float fx = px - x0f, fy = py - y0f;
            const float gx = 1.0f - fx, gy = 1.0f - fy;
            const float vx0 = (x0 >= 0 && x0 < W_)         ? 1.0f : 0.0f;
            const float vx1 = (x0 + 1 >= 0 && x0 + 1 < W_) ? 1.0f : 0.0f;
            const float vy0 = (y0 >= 0 && y0 < H_)         ? 1.0f : 0.0f;
            const float vy1 = (y0 + 1 >= 0 && y0 + 1 < H_) ? 1.0f : 0.0f;
            wgt[v][0] = gx * gy * vx0 * vy0;
            wgt[v][1] = fx * gy * vx1 * vy0;
            wgt[v][2] = gx * fy * vx0 * vy1;
            wgt[v][3] = fx * fy * vx1 * vy1;
            const int cx0 = iclampi(x0, 0, W_ - 1);
            const int cx1 = iclampi(x0 + 1, 0, W_ - 1);
            const int cy0 = iclampi(y0, 0, H_ - 1);
            const int cy1 = iclampi(y0 + 1, 0, H_ - 1);
            lin[v][0] = cy0 * W_ + cx0;
            lin[v][1] = cy0 * W_ + cx1;
            lin[v][2] = cy1 * W_ + cx0;
            lin[v][3] = cy1 * W_ + cx1;
        }

        // 10 chunks of 4 channels: cost[p] accumulated across K by WMMA.
        v8f cacc = {0.f, 0.f, 0.f, 0.f, 0.f, 0.f, 0.f, 0.f};
#pragma unroll
        for (int j = 0; j < 10; ++j) {
            float bv0 = 0.f, bv1 = 0.f;
#pragma unroll
            for (int t = 0; t < 2; ++t) {
                const int   c = 4 * j + ofs + t;
                const float r = refv[2 * j + t];
                float s = r, sq = r * r;
#pragma unroll
                for (int v = 0; v < 2; ++v) {
                    const float* src = (c < C_)
                        ? features + ((v + 1) * C_ + c) * NPIX
                        : deps + ((v + 1) * CD_ + (c - C_)) * NPIX;
                    const float val = wgt[v][0] * src[lin[v][0]] +
                                      wgt[v][1] * src[lin[v][1]] +
                                      wgt[v][2] * src[lin[v][2]] +
                                      wgt[v][3] * src[lin[v][3]];
                    s += val; sq += val * val;
                }
                const float var = sq * (1.0f / 3.0f) - s * s * (1.0f / 9.0f);
                if (t == 0) bv0 = var; else bv1 = var;
            }
            // A = reg_w chunk broadcast over rows, B = variance chunk per pixel col.
            // D[m][n] = cost[pixel n] for all m -> element 0 of D is this lane's pixel.
            const v2f a = {rwv[2 * j], rwv[2 * j + 1]};
            const v2f b = {bv0, bv1};
            cacc = __builtin_amdgcn_wmma_f32_16x16x4_f32(
                false, a, false, b, (short)0, cacc, false, false);
        }
        if (half == 0) lds_cost[slot][d] = cacc[0];
    }

    // Softmax / depth / variance: only one lane per pixel (no WMMA past this point,
    // so divergence is fine). Same thread wrote its LDS slot -> DScnt keeps order.
    if (half == 0) {
        float mx = -3.4e38f;
        for (int d = 0; d < D_; ++d) mx = fmaxf(mx, lds_cost[slot][d]);
        float ssum = 0.f;
        for (int d = 0; d < D_; ++d) {
            const float e = __expf(lds_cost[slot][d] - mx);
            lds_cost[slot][d] = e;
            ssum += e;
        }
        const float isum = 1.0f / ssum;
        float depth = 0.f;
        for (int d = 0; d < D_; ++d) {
            const float p = lds_cost[slot][d] * isum;
            lds_cost[slot][d] = p;
            depth += p * depth_values[d * NPIX + pix];
        }
        float vsum = 0.f;
        for (int d = 0; d < D_; ++d) {
            const float dv   = depth_values[d * NPIX + pix];
            const float diff = dv - depth;
            const float p    = lds_cost[slot][d];
            vsum += diff * diff * p;
            out[2 * NPIX + d * NPIX + pix] = p;       // prob_volume
        }
        out[pix]         = depth;                      // depth
        out[NPIX + pix]  = LAMB_ * sqrtf(vsum);        // exp_variance
    }
}

extern "C" void kernel_launch(void* const* d_in, const int* in_sizes, int n_in,
                              void* d_out, int out_size, void* d_ws, size_t ws_size,
                              hipStream_t stream) {
    const float* features     = (const float*)d_in[0];  // (3,1,32,128,160)
    const float* deps         = (const float*)d_in[1];  // (3,1,8,128,160)
    const float* proj         = (const float*)d_in[2];  // (1,3,4,4)
    const float* depth_values = (const float*)d_in[3];  // (1,32,128,160)
    const float* reg_w        = (const float*)d_in[4];  // (1,40)
    float*       out          = (float*)d_out;          // 20480+20480+655360
    float*       ws           = (float*)d_ws;           // needs 24 floats

    proj_setup_kernel<<<1, 1, 0, stream>>>(proj, ws);

    // 2 lanes per pixel -> 2*NPIX threads; 256 threads (8 waves) per block.
    const int threads = 256;
    const int blocks  = (2 * NPIX) / threads;           // 160
    depthnet_fused_kernel<<<blocks, threads, 0, stream>>>(
        features, deps, depth_values, reg_w, ws, out);
}